// Prompt_13365938225509
// MI455X (gfx1250) — compile-verified
//
#include <hip/hip_runtime.h>

typedef __attribute__((ext_vector_type(2))) float v2f;
typedef __attribute__((ext_vector_type(8))) float v8f;

#define B_    32
#define N_    4096
#define D_    768
#define ND_   256
#define POOL_ 1024
#define LEN_  5
#define TOPK_ 4
#define LENG_ 20
#define H_    8
#define HD_   96
#define EPS_  1e-12f
#define SCALE_ 0.10206207261596575f   /* 96^-0.5 */

#define NE_ (B_*LENG_*D_)   /* 491520 */
#define NK_ (B_*TOPK_*D_)   /*  98304 */

// ---------------------------------------------------------------------------
// Pipelined wave-level 16x16 f32 WMMA tile (attention + small GEMMs).
// A element (m,k) at A[m*arow+k]; B element (k,n) at B[k*brow+n*bcol].
// ---------------------------------------------------------------------------
__device__ __forceinline__ v8f wmma_tile(const float* A, long arow, int aRows,
                                         const float* Bm, long brow, long bcol,
                                         int K, int lane)
{
  const int half = lane >> 4;
  const int lid  = lane & 15;
  const int rm   = (lid < aRows) ? lid : (aRows - 1);
  const float* ap = A + (long)rm * arow + 2 * half;
  const float* bp = Bm + (long)(2 * half) * brow + (long)lid * bcol;
  v8f acc = {0.f, 0.f, 0.f, 0.f, 0.f, 0.f, 0.f, 0.f};

  v2f a, b;
  a.x = ap[0]; a.y = ap[1];
  b.x = bp[0]; b.y = bp[brow];
  for (int k0 = 0; k0 + 4 < K; k0 += 4) {
    v2f an, bn;                                // prefetch next K-step
    an.x = ap[k0 + 4]; an.y = ap[k0 + 5];
    bn.x = bp[(long)(k0 + 4) * brow]; bn.y = bp[(long)(k0 + 5) * brow];
    acc = __builtin_amdgcn_wmma_f32_16x16x4_f32(false, a, false, b, (short)0, acc,
                                                false, false);
    a = an; b = bn;
  }
  acc = __builtin_amdgcn_wmma_f32_16x16x4_f32(false, a, false, b, (short)0, acc,
                                              false, false);
  return acc;
}

__device__ __forceinline__ void wmma_store(float* C, long crow, v8f acc, int lane,
                                           int mValid, float bias)
{
  const int half = lane >> 4;
  const int lid  = lane & 15;
#pragma unroll
  for (int v = 0; v < 8; ++v) {
    const int m = v + 8 * half;
    if (m < mValid) C[(long)m * crow + lid] = acc[v] + bias;
  }
}

// ---------------------------------------------------------------------------
// Generic wave-per-16x16-tile GEMM + bias (small GEMMs: sim, q-proj).
// ---------------------------------------------------------------------------
__global__ void gemm16_bias_kernel(const float* __restrict__ A, long lda,
                                   const float* __restrict__ Bm, long brow, long bcol,
                                   const float* __restrict__ bias,
                                   float* __restrict__ C, long ldc,
                                   int M, int K, int tilesM, int tilesN)
{
  const int wave = threadIdx.x >> 5;
  const int lane = threadIdx.x & 31;
  const int tile = blockIdx.x * (blockDim.x >> 5) + wave;
  if (tile >= tilesM * tilesN) return;          // wave-uniform
  const int mt = tile / tilesN, nt = tile % tilesN;
  const int m0 = mt * 16, n0 = nt * 16;
  int aRows = M - m0; if (aRows > 16) aRows = 16;

  v8f acc = wmma_tile(A + (long)m0 * lda, lda, aRows,
                      Bm + (long)n0 * bcol, brow, bcol, K, lane);
  const float bb = bias ? bias[n0 + (lane & 15)] : 0.f;
  wmma_store(C + (long)m0 * ldc + n0, ldc, acc, lane, aRows, bb);
}

// ---------------------------------------------------------------------------
// Hot-path GEMM: 64x64 macro-tile per wave (4x4 subtiles, 16 v8f accumulators),
// specialized for row-major B with compile-time LDB (bcol==1). All 8 B loads
// per K-step use ONE base register + immediate offsets; 4 A loads are b64 at
// fixed bases. Double-buffered K-steps. 64-thread blocks + waves_per_eu(1)
// give the wave ~190 VGPRs with no spills (acc 128 + frags 32 + addrs ~20).
// ---------------------------------------------------------------------------
template <int LDB>
__global__ __launch_bounds__(64)
__attribute__((amdgpu_waves_per_eu(1)))
void gemm_tn_kernel(const float* __restrict__ A, long lda,
                    const float* __restrict__ Bm,
                    const float* __restrict__ bias,
                    float* __restrict__ C, long ldc,
                    int M, int K, int tilesM, int tilesN)
{
  const int wave = threadIdx.x >> 5;
  const int lane = threadIdx.x & 31;
  const int tile = blockIdx.x * 2 + wave;
  if (tile >= tilesM * tilesN) return;          // wave-uniform
  const int mt = tile / tilesN, nt = tile % tilesN;
  const int m0 = mt * 64, n0 = nt * 64;
  const int half = lane >> 4;
  const int lid  = lane & 15;

  const float* ap[4];
#pragma unroll
  for (int i = 0; i < 4; ++i) {
    int r = m0 + i * 16 + lid;
    if (r >= M) r = M - 1;                      // branchless row clamp
    ap[i] = A + (long)r * lda + 2 * half;
  }
  const float* bp = Bm + (long)(2 * half) * LDB + n0 + lid;

  v8f acc[16];
#pragma unroll
  for (int i = 0; i < 16; ++i) acc[i] = (v8f){0.f,0.f,0.f,0.f,0.f,0.f,0.f,0.f};

  v2f a[4], b[4];
#pragma unroll
  for (int i = 0; i < 4; ++i) { a[i].x = ap[i][0]; a[i].y = ap[i][1]; }
#pragma unroll
  for (int j = 0; j < 4; ++j) { b[j].x = bp[j * 16]; b[j].y = bp[LDB + j * 16]; }

  for (int k0 = 0; k0 + 4 < K; k0 += 4) {
    v2f an[4], bn[4];                           // prefetch next K-step (imm offsets)
#pragma unroll
    for (int i = 0; i < 4; ++i) { an[i].x = ap[i][4]; an[i].y = ap[i][5]; }
#pragma unroll
    for (int j = 0; j < 4; ++j) {
      bn[j].x = bp[4 * LDB + j * 16];
      bn[j].y = bp[5 * LDB + j * 16];
    }
#pragma unroll
    for (int i = 0; i < 4; ++i)
#pragma unroll
      for (int j = 0; j < 4; ++j)
        acc[i * 4 + j] = __builtin_amdgcn_wmma_f32_16x16x4_f32(
            false, a[i], false, b[j], (short)0, acc[i * 4 + j], false, false);
#pragma unroll
    for (int i = 0; i < 4; ++i) { a[i] = an[i]; b[i] = bn[i]; ap[i] += 4; }
    bp += 4 * LDB;
  }
#pragma unroll
  for (int i = 0; i < 4; ++i)
#pragma unroll
    for (int j = 0; j < 4; ++j)
      acc[i * 4 + j] = __builtin_amdgcn_wmma_f32_16x16x4_f32(
          false, a[i], false, b[j], (short)0, acc[i * 4 + j], false, false);

  // store 4x4 subtiles, guarded by valid rows
#pragma unroll
  for (int i = 0; i < 4; ++i) {
    int rowsValid = M - (m0 + i * 16);
    if (rowsValid > 16) rowsValid = 16;
#pragma unroll
    for (int j = 0; j < 4; ++j) {
      const float bb = bias ? bias[n0 + j * 16 + lid] : 0.f;
      float* cp = C + (long)(m0 + i * 16) * ldc + n0 + j * 16;
#pragma unroll
      for (int v = 0; v < 8; ++v) {
        const int m = v + 8 * half;
        if (m < rowsValid) cp[(long)m * ldc + lid] = acc[i * 4 + j][v] + bb;
      }
    }
  }
}

// ---------------------------------------------------------------------------
// Fused attention for one (b,h): S = softmax(q_h k^T * SCALE); o = S v.
// K/V read from global (50MB KV is L2-resident in MI455X's 192MB L2).
// ---------------------------------------------------------------------------
__global__ void attn_kernel(const float* __restrict__ q,    /* 20 x 768    */
                            const float* __restrict__ KV,   /* 8192 x 1536 */
                            float* __restrict__ o_buf)      /* 640 x 768   */
{
  __shared__ float S[32 * 256];
  __shared__ float red[256];
  const int bh = blockIdx.x;
  const int b = bh / H_, h = bh % H_;
  const int wave = threadIdx.x >> 5, lane = threadIdx.x & 31;
  const float* Kbase = KV + (long)b * ND_ * (2 * D_) + h * HD_;
  const float* Vbase = Kbase + D_;

  // Phase 1: scores (M=20 pad 32, N=256, K=96) -> LDS, scaled
  for (int t = wave; t < 32; t += 8) {
    const int mt = t & 1, nt = t >> 1;
    const int m0 = mt * 16, n0 = nt * 16;
    const int aRows = mt ? (LENG_ - 16) : 16;
    v8f acc = wmma_tile(q + (long)m0 * D_ + h * HD_, D_, aRows,
                        Kbase + (long)n0 * (2 * D_), 1, 2 * D_, HD_, lane);
    const int half = lane >> 4, lid = lane & 15;
#pragma unroll
    for (int v = 0; v < 8; ++v)
      S[(m0 + v + 8 * half) * 256 + n0 + lid] = acc[v] * SCALE_;
  }
  __syncthreads();

  // Phase 2: softmax over the 20 real rows
  for (int r = 0; r < LENG_; ++r) {
    const float x = S[r * 256 + threadIdx.x];
    red[threadIdx.x] = x; __syncthreads();
    for (int st = 128; st > 0; st >>= 1) {
      if (threadIdx.x < st) red[threadIdx.x] = fmaxf(red[threadIdx.x], red[threadIdx.x + st]);
      __syncthreads();
    }
    const float mx = red[0]; __syncthreads();
    const float e = __expf(x - mx);
    red[threadIdx.x] = e; __syncthreads();
    for (int st = 128; st > 0; st >>= 1) {
      if (threadIdx.x < st) red[threadIdx.x] += red[threadIdx.x + st];
      __syncthreads();
    }
    const float sum = red[0]; __syncthreads();
    S[r * 256 + threadIdx.x] = e / sum;
  }
  __syncthreads();

  // Phase 3: o = S(20x256) @ v(256x96)
  for (int t = wave; t < 12; t += 8) {
    const int mt = t / 6, nt = t % 6;
    const int m0 = mt * 16, n0 = nt * 16;
    const int aRows = mt ? (LENG_ - 16) : 16;
    v8f acc = wmma_tile(S + m0 * 256, 256, aRows,
                        Vbase + n0, 2 * D_, 1, ND_, lane);
    wmma_store(o_buf + ((long)b * LENG_ + m0) * D_ + h * HD_ + n0, D_, acc, lane, aRows, 0.f);
  }
}

// ---------------------------------------------------------------------------
__global__ void mean_kernel(const float* __restrict__ in, float* __restrict__ out, int R)
{
  const int chunks = D_ / 256;
  const int b = blockIdx.x / chunks, c = blockIdx.x % chunks;
  const int d = c * 256 + threadIdx.x;
  const float* p = in + (long)b * R * D_ + d;
  float s = 0.f;
#pragma unroll 4
  for (int r = 0; r < R; ++r) {
    __builtin_prefetch(p + (long)(r + 32) * D_, 0, 1); // global_prefetch_b8
    s += p[(long)r * D_];
  }
  out[b * D_ + d] = s / (float)R;
}

__global__ void l2norm_rows_kernel(const float* __restrict__ in, float* __restrict__ out)
{
  __shared__ float red[256];
  const long row = blockIdx.x;
  const float* p = in + row * D_;
  float s = 0.f;
  for (int d = threadIdx.x; d < D_; d += 256) { const float v = p[d]; s += v * v; }
  red[threadIdx.x] = s; __syncthreads();
  for (int st = 128; st > 0; st >>= 1) {
    if (threadIdx.x < st) red[threadIdx.x] += red[threadIdx.x + st];
    __syncthreads();
  }
  const float inv = rsqrtf(fmaxf(red[0], EPS_));
  for (int d = threadIdx.x; d < D_; d += 256) out[row * D_ + d] = p[d] * inv;
}

__global__ void topk_kernel(const float* __restrict__ sim, const float* __restrict__ freq,
                            int* __restrict__ idxOut)
{
  __shared__ float vals[POOL_];
  __shared__ float rv[256];
  __shared__ int   ri[256];
  __shared__ int   winner;
  const int b = blockIdx.x, t = threadIdx.x;

  float fm = -1e30f;
  for (int p = t; p < POOL_; p += 256) fm = fmaxf(fm, freq[p]);
  rv[t] = fm; __syncthreads();
  for (int st = 128; st > 0; st >>= 1) {
    if (t < st) rv[t] = fmaxf(rv[t], rv[t + st]);
    __syncthreads();
  }
  const float fmax_all = rv[0]; __syncthreads();

  for (int p = t; p < POOL_; p += 256)
    vals[p] = sim[b * POOL_ + p] - freq[p] / fmax_all;
  __syncthreads();

  for (int k = 0; k < TOPK_; ++k) {
    float bv = -1e38f; int bi = POOL_;
    for (int p = t; p < POOL_; p += 256) {
      const float v = vals[p];
      if (v > bv) { bv = v; bi = p; }   // ascending p -> lowest index on ties
    }
    rv[t] = bv; ri[t] = bi; __syncthreads();
    for (int st = 128; st > 0; st >>= 1) {
      if (t < st) {
        if (rv[t + st] > rv[t] || (rv[t + st] == rv[t] && ri[t + st] < ri[t])) {
          rv[t] = rv[t + st]; ri[t] = ri[t + st];
        }
      }
      __syncthreads();
    }
    if (t == 0) { winner = ri[0]; idxOut[b * TOPK_ + k] = winner; }
    __syncthreads();
    if (t == 0) vals[winner] = -1e38f;
    __syncthreads();
  }
}

__global__ void gather_kernel(const float* __restrict__ prompt,
                              const float* __restrict__ pnorm,
                              const int* __restrict__ idx,
                              float* __restrict__ out)
{
  const int i = blockIdx.x * blockDim.x + threadIdx.x;
  if (i < NE_) {
    const int b = i / (LENG_ * D_);
    const int rem = i % (LENG_ * D_);
    const int tt = rem / D_, d = rem % D_;
    const int pi = idx[b * TOPK_ + tt / LEN_];
    out[NE_ + i] = prompt[((long)pi * LEN_ + (tt % LEN_)) * D_ + d];
  } else if (i < NE_ + NK_) {
    const int i2 = i - NE_;
    const int b = i2 / (TOPK_ * D_);
    const int rem = i2 % (TOPK_ * D_);
    const int pi = idx[b * TOPK_ + rem / D_];
    out[2 * NE_ + i2] = pnorm[(long)pi * D_ + (rem % D_)];
  }
}

__global__ void reduce_sim_kernel(const float* __restrict__ g_norm,
                                  const float* __restrict__ dep_norm,
                                  const float* __restrict__ bkn,
                                  float* __restrict__ out)
{
  __shared__ float red[256];
  const int t = threadIdx.x;
  float v = 0.f;
  if (t < B_) {
    float s = 0.f;
    for (int d = 0; d < D_; ++d) s += g_norm[d] * dep_norm[t * D_ + d];
    const float e = 1.f - s; v = e * e;
  } else if (t < B_ + B_ * TOPK_) {
    const int j = t - B_;
    float s = 0.f;
    const float* p = bkn + (long)j * D_;
    for (int d = 0; d < D_; ++d) s += g_norm[d] * p[d];
    const float e = 1.f - s; v = -e * e;
  }
  red[t] = v; __syncthreads();
  for (int st = 128; st > 0; st >>= 1) {
    if (t < st) red[t] += red[t + st];
    __syncthreads();
  }
  if (t == 0) out[0] = red[0] / (float)B_;
}

// ---------------------------------------------------------------------------
extern "C" void kernel_launch(void* const* d_in, const int* in_sizes, int n_in,
                              void* d_out, int out_size, void* d_ws, size_t ws_size,
                              hipStream_t stream)
{
  (void)in_sizes; (void)n_in; (void)out_size; (void)ws_size;
  const float* x_embed   = (const float*)d_in[0];
  const float* depth     = (const float*)d_in[1];
  const float* prompt    = (const float*)d_in[2];
  const float* pkey      = (const float*)d_in[3];
  const float* pkey_g    = (const float*)d_in[4];
  const float* g_prompt  = (const float*)d_in[5];
  const float* freq      = (const float*)d_in[6];
  const float* Wq        = (const float*)d_in[7];
  const float* bq        = (const float*)d_in[8];
  const float* Wkv       = (const float*)d_in[9];
  const float* bkv       = (const float*)d_in[10];
  const float* Wproj     = (const float*)d_in[11];
  const float* bproj     = (const float*)d_in[12];
  float* out = (float*)d_out;

  float* ws       = (float*)d_ws;
  float* x_mean   = ws;
  float* dep_mean = ws + 24576;
  float* x_norm   = ws + 49152;
  float* dep_norm = ws + 73728;
  float* g_norm   = ws + 98304;
  float* pnorm    = ws + 99072;         // 786432
  float* sim      = ws + 885504;        // 32768
  int*   idx      = (int*)(ws + 918272);// 128
  float* q        = ws + 918400;        // 15360
  float* o_buf    = ws + 933760;        // 491520
  float* KV       = ws + 1425280;       // 12582912 (50 MB; L2-resident)

  mean_kernel<<<B_ * (D_ / 256), 256, 0, stream>>>(x_embed, x_mean, N_);
  mean_kernel<<<B_ * (D_ / 256), 256, 0, stream>>>(depth, dep_mean, ND_);

  l2norm_rows_kernel<<<POOL_, 256, 0, stream>>>(pkey, pnorm);
  l2norm_rows_kernel<<<B_,    256, 0, stream>>>(x_mean, x_norm);
  l2norm_rows_kernel<<<B_,    256, 0, stream>>>(dep_mean, dep_norm);
  l2norm_rows_kernel<<<1,     256, 0, stream>>>(pkey_g, g_norm);

  // sim = x_norm(32x768) @ pnorm^T (transposed B via strides): 2x64 16-tiles
  gemm16_bias_kernel<<<16, 256, 0, stream>>>(x_norm, D_, pnorm, 1, D_, nullptr,
                                             sim, POOL_, B_, D_, 2, 64);
  topk_kernel<<<B_, 256, 0, stream>>>(sim, freq, idx);

  // q = g_prompt(20x768) @ Wq + bq (batch-invariant, computed once): 2x48 tiles
  gemm16_bias_kernel<<<12, 256, 0, stream>>>(g_prompt, D_, Wq, D_, 1, bq,
                                             q, D_, LENG_, D_, 2, 48);
  // KV = depth(8192x768) @ Wkv(768x1536) + bkv (19.3 GFLOP; 128x24 macro-tiles,
  // 2 waves/block, 3072 waves)
  gemm_tn_kernel<2 * D_><<<1536, 64, 0, stream>>>(depth, D_, Wkv, bkv,
                                                  KV, 2 * D_, B_ * ND_, D_, 128, 24);
  attn_kernel<<<B_ * H_, 256, 0, stream>>>(q, KV, o_buf);
  // prompted_g = o_buf(640x768) @ Wproj + bproj -> out[0:NE_)  (10x12 macro-tiles)
  gemm_tn_kernel<D_><<<60, 64, 0, stream>>>(o_buf, D_, Wproj, bproj,
                                            out, D_, B_ * LENG_, D_, 10, 12);
  gather_kernel<<<(NE_ + NK_) / 256, 256, 0, stream>>>(prompt, pnorm, idx, out);
  reduce_sim_kernel<<<1, 256, 0, stream>>>(g_norm, dep_norm, out + 2 * NE_,
                                           out + 2 * NE_ + NK_);
}